// RConv_58961311039764
// MI455X (gfx1250) — compile-verified
//
#include <hip/hip_runtime.h>

typedef float v2f __attribute__((ext_vector_type(2)));
typedef float v8f __attribute__((ext_vector_type(8)));

#define W_IMG 192
#define H_IMG 192
#define C_CH  64
#define HW    (W_IMG * H_IMG)
#define ROWS_PER_BLK 8
#define STAGE_ROWS   (ROWS_PER_BLK + 2)   // halo rows
#define STAGE_COLS   18                   // 16 + 2 halo cols
#define IC_STRIDE    208                  // words per ic; ≡16 mod 32 so half-wave lands +32 banks
#define N_TAPS 5
#define NW_ELEMS (N_TAPS * 16 * 4 * 64)   // 20480 floats = 80 KB
#define NX_ELEMS (C_CH * IC_STRIDE)       // 13312 floats = 52 KB

__global__ __launch_bounds__(128) void rconv_wmma_f32(
    const float* __restrict__ x,      // [16,64,192,192]
    const float* __restrict__ w,      // [64,64,5]
    const int*   __restrict__ rotp,   // scalar
    float* __restrict__ out)          // [16,64,192,192]
{
    __shared__ float lds_w[NW_ELEMS]; // A tiles: [tap][chunk][ocg][lane*2+q]
    __shared__ float lds_x[NX_ELEMS]; // input:   [ic][row][col] (padded ic stride)

    const int tid  = threadIdx.x;
    const int lane = tid & 31;
    const int wv   = tid >> 5;
    const int half = lane >> 4;   // 0: lanes 0-15, 1: lanes 16-31
    const int l16  = lane & 15;

    const int xb = blockIdx.x * 16;
    const int y0 = blockIdx.y * ROWS_PER_BLK;
    const int n  = blockIdx.z;

    // ---- stage weights into WMMA A-tile lane layout ----
    // f = q + 2*L + 64*g + 256*c + 4096*t  ->  lds_w[((t*16+c)*4+g)*64 + L*2 + q]
    for (int f = tid; f < NW_ELEMS; f += 128) {
        int q = f & 1;
        int L = (f >> 1) & 31;
        int g = (f >> 6) & 3;
        int c = (f >> 8) & 15;
        int t = f >> 12;
        int o = g * 16 + (L & 15);                 // M (oc within group)
        int i = c * 4 + ((L >> 4) << 1) + q;       // K (ic within chunk)
        lds_w[f] = w[(o * C_CH + i) * N_TAPS + t];
    }

    // ---- stage input halo tile (zero-padded at image edges) ----
    const float* xin = x + (size_t)n * C_CH * HW;
    for (int f = tid; f < C_CH * STAGE_ROWS * STAGE_COLS; f += 128) {
        int col = f % STAGE_COLS;
        int tmp = f / STAGE_COLS;
        int row = tmp % STAGE_ROWS;
        int ic  = tmp / STAGE_ROWS;
        int gy  = y0 - 1 + row;
        int gx  = xb - 1 + col;
        float v = 0.0f;
        if ((unsigned)gy < (unsigned)H_IMG && (unsigned)gx < (unsigned)W_IMG)
            v = xin[ic * HW + gy * W_IMG + gx];
        lds_x[ic * IC_STRIDE + row * STAGE_COLS + col] = v;
    }
    __syncthreads();

    // ---- tap positions from runtime rot (skel kernel: 4 ring taps + center) ----
    const int ringr[8] = {0, 0, 1, 2, 2, 2, 1, 0};
    const int ringc[8] = {1, 2, 2, 2, 1, 0, 0, 0};
    int off = ((rotp[0] % 8) + 8) % 8;
    int posr[N_TAPS], posc[N_TAPS];
#pragma unroll
    for (int k = 0; k < 4; ++k) {
        int p = (2 * k + off) & 7;
        posr[k] = ringr[p];
        posc[k] = ringc[p];
    }
    posr[4] = 1; posc[4] = 1;

    // ---- WMMA main loop: wave handles rows [2*wv, 2*wv+1], all 64 oc ----
    v8f acc[2][4];
#pragma unroll
    for (int lr = 0; lr < 2; ++lr)
#pragma unroll
        for (int g = 0; g < 4; ++g)
#pragma unroll
            for (int j = 0; j < 8; ++j)
                acc[lr][g][j] = 0.0f;

    const int r0   = wv * 2;
    const int ksel = half * 2;   // B lane -> K row (mirror of A layout: K0/K1 in lanes 0-15, K2/K3 in lanes 16-31)

#pragma unroll
    for (int t = 0; t < N_TAPS; ++t) {
        const int pr = posr[t];
        const int pc = posc[t];
        for (int c = 0; c < 16; ++c) {
            v2f a[4];
#pragma unroll
            for (int g = 0; g < 4; ++g)
                a[g] = *(const v2f*)&lds_w[((t * 16 + c) * 4 + g) * 64 + lane * 2];
#pragma unroll
            for (int lr = 0; lr < 2; ++lr) {
                const int srow = r0 + lr + pr;
                const int scol = pc + l16;
                const int ic0  = c * 4 + ksel;
                v2f b;
                b.x = lds_x[ ic0      * IC_STRIDE + srow * STAGE_COLS + scol];
                b.y = lds_x[(ic0 + 1) * IC_STRIDE + srow * STAGE_COLS + scol];
#pragma unroll
                for (int g = 0; g < 4; ++g)
                    acc[lr][g] = __builtin_amdgcn_wmma_f32_16x16x4_f32(
                        false, a[g], false, b, (short)0, acc[lr][g], false, false);
            }
        }
    }

    // ---- epilogue: D layout (ISA §7.12.2): VGPR j, lane L -> M = j + half*8, N = l16 ----
#pragma unroll
    for (int lr = 0; lr < 2; ++lr) {
        const int y = y0 + r0 + lr;
#pragma unroll
        for (int g = 0; g < 4; ++g) {
#pragma unroll
            for (int j = 0; j < 8; ++j) {
                const int oc = g * 16 + j + half * 8;
                out[(((size_t)n * C_CH + oc) * H_IMG + y) * W_IMG + xb + l16] = acc[lr][g][j];
            }
        }
    }
}

extern "C" void kernel_launch(void* const* d_in, const int* in_sizes, int n_in,
                              void* d_out, int out_size, void* d_ws, size_t ws_size,
                              hipStream_t stream) {
    const float* x    = (const float*)d_in[0];
    const float* wgt  = (const float*)d_in[1];
    const int*   rotp = (const int*)d_in[2];
    float* out = (float*)d_out;

    dim3 grid(W_IMG / 16, H_IMG / ROWS_PER_BLK, 16);  // 12 x 24 x 16
    dim3 block(128);
    rconv_wmma_f32<<<grid, block, 0, stream>>>(x, wgt, rotp, out);
}